// CustomAttention_54116587930137
// MI455X (gfx1250) — compile-verified
//
#include <hip/hip_runtime.h>
#include <hip/hip_bf16.h>

typedef __attribute__((ext_vector_type(2))) float v2f;
typedef __attribute__((ext_vector_type(8))) float v8f;

#define DSIZE 1024
#define BATCH 32
#define SEQ   2048

// ---------------------------------------------------------------------------
// Kernel A: v[b,d] = sum_e dh[b,e] * W[e,d]   (M=32, N=1024, K=1024 f32 GEMM)
// One wave (32 threads) per 16x16 output tile; K-loop of V_WMMA_F32_16X16X4_F32.
// VGPR layouts per CDNA5 ISA 7.12.2:
//   A (16x4):  VGPR0 = K=0 (lanes 0-15) / K=2 (lanes 16-31), VGPR1 = K=1 / K=3
//   B (4x16):  VGPR0 = row K=0 (lanes 0-15) / row K=2 (lanes 16-31), VGPR1 = K=1 / K=3
//   C/D:       VGPR r = M=r (lanes 0-15) / M=r+8 (lanes 16-31), N = lane&15
// ---------------------------------------------------------------------------
__global__ __launch_bounds__(32) void dhW_wmma_kernel(const float* __restrict__ dh,
                                                      const float* __restrict__ W,
                                                      float* __restrict__ v) {
    const int lane  = threadIdx.x;      // 0..31, EXEC all ones (no divergence)
    const int half  = lane >> 4;        // 0 or 1
    const int l15   = lane & 15;
    const int nBase = blockIdx.x * 16;  // d tile
    const int mBase = blockIdx.y * 16;  // b tile

    const float* aRow = dh + (mBase + l15) * DSIZE;
    v8f acc = {};

    for (int k = 0; k < DSIZE; k += 4) {
        const int kk = k + 2 * half;
        v2f a, b;
        a.x = aRow[kk];
        a.y = aRow[kk + 1];
        b.x = W[(kk + 0) * DSIZE + nBase + l15];
        b.y = W[(kk + 1) * DSIZE + nBase + l15];
        acc = __builtin_amdgcn_wmma_f32_16x16x4_f32(
            /*neg_a=*/false, a, /*neg_b=*/false, b,
            /*c_mod=*/(short)0, acc, /*reuse_a=*/false, /*reuse_b=*/false);
    }

#pragma unroll
    for (int r = 0; r < 8; ++r) {
        v[(mBase + r + 8 * half) * DSIZE + nBase + l15] = acc[r];
    }
}

// ---------------------------------------------------------------------------
// Kernel B: E[b,s] = dot(enc[s,b,:], v[b,:])  — bandwidth-bound streaming.
// Grid: (SEQ/8, BATCH); block = 256 threads = 8 waves; one wave per s-row.
// v_b (4 KB) staged in LDS once per block; enc rows streamed as float4,
// 512 B contiguous per wave-load => fully coalesced, each HBM byte read once.
// ---------------------------------------------------------------------------
__global__ __launch_bounds__(256) void energies_kernel(const float* __restrict__ enc,
                                                       const float* __restrict__ v,
                                                       float* __restrict__ E) {
    __shared__ __align__(16) float vb[DSIZE];
    const int b   = blockIdx.y;
    const int tid = threadIdx.x;

    // cooperative LDS fill: 256 threads x float4 = 1024 floats
    ((float4*)vb)[tid] = ((const float4*)(v + b * DSIZE))[tid];
    __syncthreads();

    const int wave = tid >> 5;
    const int lane = tid & 31;
    const int s    = blockIdx.x * 8 + wave;

    const float4* row = (const float4*)(enc + ((size_t)s * BATCH + b) * DSIZE);
    const float4* vv  = (const float4*)vb;

    float sum = 0.0f;
#pragma unroll
    for (int i = 0; i < 8; ++i) {
        const float4 e4 = row[lane + i * 32];
        const float4 w4 = vv[lane + i * 32];
        sum = fmaf(e4.x, w4.x, sum);
        sum = fmaf(e4.y, w4.y, sum);
        sum = fmaf(e4.z, w4.z, sum);
        sum = fmaf(e4.w, w4.w, sum);
    }
#pragma unroll
    for (int off = 16; off > 0; off >>= 1) sum += __shfl_xor(sum, off, 32);
    if (lane == 0) E[b * SEQ + s] = sum;
}

// ---------------------------------------------------------------------------
// Kernel C: out[b,s] = softmax_s(E[b,:]).  One block per b, 2048 elems,
// 8 per thread.  (Bias term dh.b is a per-b constant -> cancels in softmax.)
// ---------------------------------------------------------------------------
__global__ __launch_bounds__(256) void softmax_kernel(const float* __restrict__ E,
                                                      float* __restrict__ out) {
    __shared__ float red[8];
    const int b   = blockIdx.x;
    const int tid = threadIdx.x;
    const float* e = E + b * SEQ;

    float vals[8];
    float m = -INFINITY;
#pragma unroll
    for (int i = 0; i < 8; ++i) {
        vals[i] = e[tid + i * 256];
        m = fmaxf(m, vals[i]);
    }
#pragma unroll
    for (int off = 16; off > 0; off >>= 1) m = fmaxf(m, __shfl_xor(m, off, 32));
    if ((tid & 31) == 0) red[tid >> 5] = m;
    __syncthreads();
    float gmax = red[0];
#pragma unroll
    for (int i = 1; i < 8; ++i) gmax = fmaxf(gmax, red[i]);
    __syncthreads();

    float s = 0.0f;
#pragma unroll
    for (int i = 0; i < 8; ++i) {
        vals[i] = __expf(vals[i] - gmax);
        s += vals[i];
    }
#pragma unroll
    for (int off = 16; off > 0; off >>= 1) s += __shfl_xor(s, off, 32);
    if ((tid & 31) == 0) red[tid >> 5] = s;
    __syncthreads();
    float gsum = 0.0f;
#pragma unroll
    for (int i = 0; i < 8; ++i) gsum += red[i];
    const float inv = __frcp_rn(gsum);

#pragma unroll
    for (int i = 0; i < 8; ++i) out[b * SEQ + tid + i * 256] = vals[i] * inv;
}

extern "C" void kernel_launch(void* const* d_in, const int* in_sizes, int n_in,
                              void* d_out, int out_size, void* d_ws, size_t ws_size,
                              hipStream_t stream) {
    const float* dh  = (const float*)d_in[0];   // [32, 1024]
    const float* enc = (const float*)d_in[1];   // [2048, 32, 1024]
    const float* W   = (const float*)d_in[2];   // [1024, 1024]
    // d_in[3] (bias) intentionally unused: its contribution is constant per b
    // and cancels in the softmax.

    float* v = (float*)d_ws;                    // 32*1024 f32 = 128 KB
    float* E = v + BATCH * DSIZE;               // 32*2048 f32 = 256 KB
    float* o = (float*)d_out;                   // [32, 1, 2048]

    dhW_wmma_kernel<<<dim3(DSIZE / 16, BATCH / 16), 32, 0, stream>>>(dh, W, v);
    energies_kernel<<<dim3(SEQ / 8, BATCH), 256, 0, stream>>>(enc, v, E);
    softmax_kernel<<<BATCH, 256, 0, stream>>>(E, o);
}